// DetectorLoss_7438883357169
// MI455X (gfx1250) — compile-verified
//
#include <hip/hip_runtime.h>
#include <math.h>

typedef float v2f __attribute__((ext_vector_type(2)));
typedef float v8f __attribute__((ext_vector_type(8)));

#define NB   16
#define CC   85
#define HH   160
#define WW   160
#define MM   4096
#define HW_  (HH * WW)      /* 25600  */
#define NHW  (NB * HW_)     /* 409600 */
#define M4   (4 * MM)       /* 16384  */
#define EPSV 1e-7f

#define K2_BLOCKS (M4 / 256)   /* 64  */
#define K5_BLOCKS 160
#define ENC_SENTINEL (-2147483647 - 1)   /* INT_MIN: "never written" */

// ---------- order-preserving float<->int encoding (for deterministic scatter-max) ----------
__device__ __forceinline__ int enc_f32(float f) {
    int i = __float_as_int(f);
    return (i >= 0) ? i : (i ^ 0x7fffffff);
}
__device__ __forceinline__ float dec_f32(int e) {
    int i = (e >= 0) ? e : (e ^ 0x7fffffff);
    return __int_as_float(i);
}

// ---------- WMMA cross-lane wave reduction ----------
// A(16x4) holds per-lane partials (a.x = s, a.y = 0), B(4x16) = ones.
// D[m][n] = s_m + s_{m+16}; summing the 8 accumulator VGPRs gives, in lane 0,
// rows 0..7 and, in lane 16, rows 8..15 -> lane0 + lane16 = full wave sum.
__device__ __forceinline__ float wmma_wave_partial(float s) {
    v2f a; a.x = s;    a.y = 0.0f;
    v2f b; b.x = 1.0f; b.y = 1.0f;
    v8f c = {};
    c = __builtin_amdgcn_wmma_f32_16x16x4_f32(
            /*neg_a=*/false, a, /*neg_b=*/false, b,
            /*c_mod=*/(short)0, c, /*reuse_a=*/false, /*reuse_b=*/false);
    return c[0] + c[1] + c[2] + c[3] + c[4] + c[5] + c[6] + c[7];
}

// Deterministic block sum (fixed combine order). Valid in thread 0 only.
// Requires: all threads of the (256-thread) block reach this call (EXEC all 1s).
__device__ __forceinline__ float block_sum(float s, float* lds) {
    float t = wmma_wave_partial(s);
    int lane = threadIdx.x & 31;
    int wave = threadIdx.x >> 5;
    if (lane == 0)       lds[2 * wave]     = t;
    else if (lane == 16) lds[2 * wave + 1] = t;
    __syncthreads();
    float r = 0.0f;
    if (threadIdx.x == 0) {
#pragma unroll
        for (int i = 0; i < 16; ++i) r += lds[i];   // fixed order
    }
    __syncthreads();   // safe LDS reuse by a following block_sum
    return r;
}

__device__ __forceinline__ float sigmoidf_(float x) {
    return 1.0f / (1.0f + __expf(-x));
}

// ---------- SIoU ----------
__device__ float siou(float p0, float p1, float p2, float p3,
                      float g0, float g1, float g2, float g3) {
    float b1x1 = p0 - p2 * 0.5f, b1x2 = p0 + p2 * 0.5f;
    float b1y1 = p1 - p3 * 0.5f, b1y2 = p1 + p3 * 0.5f;
    float b2x1 = g0 - g2 * 0.5f, b2x2 = g0 + g2 * 0.5f;
    float b2y1 = g1 - g3 * 0.5f, b2y2 = g1 + g3 * 0.5f;

    float iw = fminf(b1x2, b2x2) - fmaxf(b1x1, b2x1);
    float ih = fminf(b1y2, b2y2) - fmaxf(b1y1, b2y1);
    float inter = fmaxf(iw, 0.0f) * fmaxf(ih, 0.0f);

    float w1 = b1x2 - b1x1, h1 = b1y2 - b1y1 + EPSV;
    float w2 = b2x2 - b2x1, h2 = b2y2 - b2y1 + EPSV;
    float uni = w1 * h1 + w2 * h2 - inter + EPSV;
    float iou = inter / uni;

    float cw = fmaxf(b1x2, b2x2) - fminf(b1x1, b2x1);
    float ch = fmaxf(b1y2, b2y2) - fminf(b1y1, b2y1);
    float s_cw = (b2x1 + b2x2 - b1x1 - b1x2) * 0.5f;
    float s_ch = (b2y1 + b2y2 - b1y1 - b1y2) * 0.5f;
    float sigma = sqrtf(s_cw * s_cw + s_ch * s_ch);
    float sin_a1 = fabsf(s_cw) / sigma;
    float sin_a2 = fabsf(s_ch) / sigma;
    const float thr = 0.70710678f;
    float sin_a = (sin_a1 > thr) ? sin_a2 : sin_a1;
    // cos(2*asin(x) - pi/2) == 2*x*sqrt(1-x^2)
    float angle_cost = 2.0f * sin_a * sqrtf(fmaxf(1.0f - sin_a * sin_a, 0.0f));

    float rx = s_cw / cw, ry = s_ch / ch;
    float rho_x = rx * rx, rho_y = ry * ry;
    float gamma = angle_cost - 2.0f;
    float distance_cost = 2.0f - __expf(gamma * rho_x) - __expf(gamma * rho_y);

    float ow = fabsf(w1 - w2) / fmaxf(w1, w2);
    float oh = fabsf(h1 - h2) / fmaxf(h1, h2);
    float ew = 1.0f - __expf(-ow), eh = 1.0f - __expf(-oh);
    float ew2 = ew * ew, eh2 = eh * eh;
    float shape_cost = ew2 * ew2 + eh2 * eh2;

    return iou - 0.5f * (distance_cost + shape_cost);
}

// ---------- K1: init tobj_enc / factor / nperb ----------
__global__ __launch_bounds__(256) void k1_init(int* tobj_enc, float* factor, int* nperb) {
    int i = blockIdx.x * blockDim.x + threadIdx.x;
    if (i < NHW) {
        tobj_enc[i] = ENC_SENTINEL;
        factor[i]   = 0.75f;
    }
    if (i < NB) nperb[i] = 0;
}

// ---------- K2: per-candidate SIoU + NLL; deterministic block partials ----------
__global__ __launch_bounds__(256) void k2_cand(const float* __restrict__ preds,
                                               const float* __restrict__ targets,
                                               float* iouA, float* nllA,
                                               int* idxA, int* bA, int* mA,
                                               float* p_cm, float* p_iou) {
    __shared__ float lds[16];
    int cand = blockIdx.x * blockDim.x + threadIdx.x;   // exactly M4 threads
    int q = cand / MM;
    int t = cand - q * MM;
    const float* tv = targets + (size_t)t * 6;
    float cx = tv[2] * (float)WW;
    float cy = tv[3] * (float)HH;
    float gw = tv[4] * (float)WW;
    float gh = tv[5] * (float)HH;
    int qx = q & 1, qy = q >> 1;
    int gij_x = (int)cx + qx;
    int gij_y = (int)cy + qy;
    int m = (gij_x > 0) && (gij_x < HH) && (gij_y > 0) && (gij_y < HH);
    int gi = m ? gij_x : 0;
    int gj = m ? gij_y : 0;
    int b = (int)tv[0];
    int gcls = (int)tv[1];

    size_t base = (size_t)b * CC * HW_ + (size_t)gj * WW + gi;
    float pr0 = preds[base + (size_t)1 * HW_];
    float pr1 = preds[base + (size_t)2 * HW_];
    float pr2 = preds[base + (size_t)3 * HW_];
    float pr3 = preds[base + (size_t)4 * HW_];

    float pt0 = tanhf(pr0) + (float)gi;
    float pt1 = tanhf(pr1) + (float)gj;
    float pt2 = sigmoidf_(pr2) * (float)WW;
    float pt3 = sigmoidf_(pr3) * (float)HH;

    float iou = siou(pt0, pt1, pt2, pt3, cx, cy, gw, gh);
    float nll = -__logf(preds[base + (size_t)(5 + gcls) * HW_]);

    iouA[cand] = iou;
    nllA[cand] = nll;
    idxA[cand] = b * HW_ + gj * WW + gi;
    bA[cand]   = b;
    mA[cand]   = m;

    // WMMA-based deterministic block reductions (all 256 threads participate)
    float bm = block_sum(m ? 1.0f : 0.0f, lds);   // count is exact in f32 (<= 256)
    float bi = block_sum(m ? iou : 0.0f, lds);
    if (threadIdx.x == 0) {
        p_cm[blockIdx.x]  = bm;
        p_iou[blockIdx.x] = bi;
    }
}

// ---------- K2b: iou_mean (single thread, fixed order) ----------
__global__ void k2b_mean(const float* __restrict__ p_cm,
                         const float* __restrict__ p_iou, float* sc) {
    if (threadIdx.x == 0 && blockIdx.x == 0) {
        float cm = 0.0f, si = 0.0f;
        for (int i = 0; i < K2_BLOCKS; ++i) { cm += p_cm[i]; si += p_iou[i]; }
        sc[0] = si / fmaxf(cm, 1.0f);   // iou_mean
    }
}

// ---------- K3: filter f, deterministic scatter-max tobj, nperb, partials ----------
__global__ __launch_bounds__(256) void k3_filter(const float* __restrict__ iouA,
                                                 const float* __restrict__ nllA,
                                                 const int* __restrict__ idxA,
                                                 const int* __restrict__ bA,
                                                 const int* __restrict__ mA,
                                                 const float* __restrict__ sc,
                                                 int* fA, int* tobj_enc, int* nperb,
                                                 float* p_cf, float* p_s1, float* p_nll) {
    __shared__ float lds[16];
    int cand = blockIdx.x * blockDim.x + threadIdx.x;
    float iou_mean = sc[0];

    float iou = iouA[cand];
    int f = mA[cand] && (iou > iou_mean);
    fA[cand] = f;
    if (f) {
        atomicMax(&tobj_enc[idxA[cand]], enc_f32(iou));  // order-independent
        atomicAdd(&nperb[bA[cand]], 1);                  // integer: deterministic
    }
    float bf = block_sum(f ? 1.0f : 0.0f, lds);
    float b1 = block_sum(f ? (1.0f - iou) : 0.0f, lds);
    float bn = block_sum(f ? nllA[cand] : 0.0f, lds);
    if (threadIdx.x == 0) {
        p_cf[blockIdx.x]  = bf;
        p_s1[blockIdx.x]  = b1;
        p_nll[blockIdx.x] = bn;
    }
}

// ---------- K4: factor scatter (duplicates write identical values) ----------
__global__ __launch_bounds__(256) void k4_factor(const int* __restrict__ fA,
                                                 const int* __restrict__ idxA,
                                                 const int* __restrict__ bA,
                                                 const int* __restrict__ nperb,
                                                 float* factor) {
    int cand = blockIdx.x * blockDim.x + threadIdx.x;
    if (fA[cand]) {
        float np = fmaxf((float)nperb[bA[cand]], 1.0f);
        factor[idxA[cand]] = (float)HW_ / np * 0.25f;
    }
}

// ---------- K5: obj smooth-L1 reduction; WMMA wave reduce + fixed block combine ----------
__global__ __launch_bounds__(256) void k5_obj(const float* __restrict__ preds,
                                              const int* __restrict__ tobj_enc,
                                              const float* __restrict__ factor,
                                              float* p_obj) {
    __shared__ float lds[16];
    int tid = blockIdx.x * blockDim.x + threadIdx.x;
    int stride = gridDim.x * blockDim.x;   // 40960: NHW/stride == 10 exactly
    float s = 0.0f;
    for (int i = tid; i < NHW; i += stride) {
        int n = i / HW_;
        int r = i - n * HW_;
        float pobj = preds[(size_t)n * CC * HW_ + r];   // channel 0
        int e = tobj_enc[i];
        float tobj = (e == ENC_SENTINEL) ? 0.0f : dec_f32(e);
        float diff = pobj - tobj;
        float ad = fabsf(diff);
        float sl1 = (ad < 1.0f) ? 0.5f * diff * diff : ad - 0.5f;
        s += sl1 * factor[i];
    }
    float bs = block_sum(s, lds);
    if (threadIdx.x == 0) p_obj[blockIdx.x] = bs;
}

// ---------- K6: finalize (single thread, fixed order) ----------
__global__ void k6_final(const float* __restrict__ p_cf,
                         const float* __restrict__ p_s1,
                         const float* __restrict__ p_nll,
                         const float* __restrict__ p_obj,
                         float* out) {
    if (threadIdx.x == 0 && blockIdx.x == 0) {
        float cf = 0.0f, s1 = 0.0f, sn = 0.0f, so = 0.0f;
        for (int i = 0; i < K2_BLOCKS; ++i) { cf += p_cf[i]; s1 += p_s1[i]; sn += p_nll[i]; }
        for (int i = 0; i < K5_BLOCKS; ++i) { so += p_obj[i]; }
        float cnt_f = fmaxf(cf, 1.0f);
        float iou_loss = s1 / cnt_f;
        float cls_loss = sn / cnt_f;
        float obj_loss = so / (float)NHW;
        out[0] = iou_loss;
        out[1] = obj_loss;
        out[2] = cls_loss;
        out[3] = iou_loss * 8.0f + obj_loss * 16.0f + cls_loss;
    }
}

extern "C" void kernel_launch(void* const* d_in, const int* in_sizes, int n_in,
                              void* d_out, int out_size, void* d_ws, size_t ws_size,
                              hipStream_t stream) {
    const float* preds   = (const float*)d_in[0];   // (16,85,160,160) f32
    const float* targets = (const float*)d_in[1];   // (4096,6) f32
    float* out = (float*)d_out;                     // 4 scalars

    // workspace carve-up (~3.7 MB)
    int*   tobj_enc = (int*)d_ws;                   // NHW
    float* factor   = (float*)(tobj_enc + NHW);     // NHW
    float* iouA     = factor + NHW;                 // M4
    float* nllA     = iouA + M4;                    // M4
    int*   idxA     = (int*)(nllA + M4);            // M4
    int*   bA       = idxA + M4;                    // M4
    int*   mA       = bA + M4;                      // M4
    int*   fA       = mA + M4;                      // M4
    float* p_cm     = (float*)(fA + M4);            // 64
    float* p_iou    = p_cm + K2_BLOCKS;             // 64
    float* p_cf     = p_iou + K2_BLOCKS;            // 64
    float* p_s1     = p_cf + K2_BLOCKS;             // 64
    float* p_nll    = p_s1 + K2_BLOCKS;             // 64
    float* p_obj    = p_nll + K2_BLOCKS;            // 160
    float* sc       = p_obj + K5_BLOCKS;            // 4 scalars
    int*   nperb    = (int*)(sc + 4);               // 16

    k1_init  <<<(NHW + 255) / 256, 256, 0, stream>>>(tobj_enc, factor, nperb);
    k2_cand  <<<K2_BLOCKS, 256, 0, stream>>>(preds, targets, iouA, nllA, idxA, bA, mA,
                                             p_cm, p_iou);
    k2b_mean <<<1, 32, 0, stream>>>(p_cm, p_iou, sc);
    k3_filter<<<K2_BLOCKS, 256, 0, stream>>>(iouA, nllA, idxA, bA, mA, sc,
                                             fA, tobj_enc, nperb, p_cf, p_s1, p_nll);
    k4_factor<<<K2_BLOCKS, 256, 0, stream>>>(fA, idxA, bA, nperb, factor);
    k5_obj   <<<K5_BLOCKS, 256, 0, stream>>>(preds, tobj_enc, factor, p_obj);
    k6_final <<<1, 32, 0, stream>>>(p_cf, p_s1, p_nll, p_obj, out);
}